// LEConv_83992380440997
// MI455X (gfx1250) — compile-verified
//
#include <hip/hip_runtime.h>
#include <hip/hip_bf16.h>

typedef __attribute__((ext_vector_type(2))) float v2f;
typedef __attribute__((ext_vector_type(8))) float v8f;

#define CH 64

// Map matrix element B[k][n] (k,n in 0..63) to its position in the
// pre-swizzled LDS B-fragment buffer: fragment (s = k/4, t = n/16) holds,
// per lane (hi*16+ln), the float2 {B[4s+2hi][t*16+ln], B[4s+2hi+1][...]}.
__device__ __forceinline__ int bfrag_idx(int k, int n) {
    const int s  = k >> 2;
    const int j  = k & 1;
    const int hi = (k >> 1) & 1;
    const int t  = n >> 4;
    const int ln = n & 15;
    return (((s * 4 + t) * 32) + hi * 16 + ln) * 2 + j;
}

// ---------------------------------------------------------------------------
// Kernel 0: zero the aggregation target (d_out) and the degree buffer.
// ---------------------------------------------------------------------------
__global__ __launch_bounds__(256) void kzero(float* __restrict__ out,
                                             float* __restrict__ deg, int m) {
    int i = blockIdx.x * 256 + threadIdx.x;
    int total = m * CH;
    if (i < total) out[i] = 0.0f;
    if (i < m) deg[i] = 0.0f;
}

// ---------------------------------------------------------------------------
// Kernel 1: emb_w = X @ weight   ([nrows,64] x [64,64], fp32 WMMA 16x16x4)
// One wave per 16-row tile; 4 column tiles of 16; 16 K-steps of 4.
// Weight staged in LDS pre-swizzled to the WMMA B-fragment layout so each
// WMMA consumes one naturally-aligned ds_load_b64 (no repacking movs).
// ---------------------------------------------------------------------------
__global__ __launch_bounds__(256) void kproj(const float* __restrict__ X,
                                             const float* __restrict__ W,
                                             float* __restrict__ embw,
                                             int nrows) {
    __shared__ __align__(16) float ldsW[CH * CH];
    for (int i = threadIdx.x; i < CH * CH; i += 256)
        ldsW[bfrag_idx(i >> 6, i & 63)] = W[i];   // W is [k][n] row-major
    __syncthreads();

    const int wave = threadIdx.x >> 5;
    const int lane = threadIdx.x & 31;
    const int hi   = lane >> 4;       // 0: K 0..1 half, 1: K 2..3 half
    const int ln   = lane & 15;

    const int rowBase = (blockIdx.x * 8 + wave) * 16;
    if (rowBase >= nrows) return;     // wave-uniform: EXEC stays all-ones

    int row = rowBase + ln;
    if (row >= nrows) row = nrows - 1;                 // clamp (loads only)
    const float* arow = X + (size_t)row * CH + 2 * hi; // per-lane A stream
    __builtin_prefetch(arow, 0, 0);

    const float* bbase = ldsW + (lane << 1);

    v8f acc[4] = {};
#pragma unroll
    for (int s = 0; s < 16; ++s) {
        v2f a;
        a.x = arow[4 * s + 0];        // A[M=row][K=4s+2*hi]
        a.y = arow[4 * s + 1];        // A[M=row][K=4s+2*hi+1]
#pragma unroll
        for (int t = 0; t < 4; ++t) {
            const v2f b = *(const v2f*)(bbase + ((s * 4 + t) << 6));
            acc[t] = __builtin_amdgcn_wmma_f32_16x16x4_f32(
                false, a, false, b, (short)0, acc[t], false, false);
        }
    }
    // D layout: lane L, VGPR r -> D[M = r + 8*(L>=16)][N = L&15] (+col tile)
#pragma unroll
    for (int t = 0; t < 4; ++t) {
#pragma unroll
        for (int r = 0; r < 8; ++r) {
            const int orow = rowBase + r + 8 * hi;
            if (orow < nrows)
                embw[(size_t)orow * CH + t * 16 + ln] = acc[t][r];
        }
    }
}

// ---------------------------------------------------------------------------
// Kernel 2: edge gather + scatter-add.
//   aggr[index1[e]] += emb_w[index[e]];  deg[index1[e]] += edge_weight[e]
// 16 threads per edge, float4 per thread (emb_w is L2-resident: 12.8 MB).
// ---------------------------------------------------------------------------
__global__ __launch_bounds__(256) void kscatter(const int* __restrict__ index,
                                                const int* __restrict__ index1,
                                                const float* __restrict__ ew,
                                                const float* __restrict__ embw,
                                                float* __restrict__ out,
                                                float* __restrict__ deg,
                                                int e) {
    const long long tid = (long long)blockIdx.x * 256 + threadIdx.x;
    const int edge = (int)(tid >> 4);
    if (edge >= e) return;
    const int c   = (int)(tid & 15) * 4;
    const int seg = index1[edge];
    const int src = index[edge];
    const float4 v = *(const float4*)(embw + (size_t)src * CH + c);
    float* dst = out + (size_t)seg * CH + c;
    atomicAdd(dst + 0, v.x);
    atomicAdd(dst + 1, v.y);
    atomicAdd(dst + 2, v.z);
    atomicAdd(dst + 3, v.w);
    if ((tid & 15) == 0) atomicAdd(deg + seg, ew[edge]);
}

// ---------------------------------------------------------------------------
// Kernel 3: x = X[valid]; out = deg*(x@W1^T + b1) + aggr(out) + x@W2^T + b2
// Two fp32 WMMA GEMMs per tile; W1/W2 staged in LDS pre-swizzled into the
// B-fragment layout (B[k][n] = W[n][k] handled by the same swizzle map).
// ---------------------------------------------------------------------------
__global__ __launch_bounds__(256) void kfinal(const float* __restrict__ X,
                                              const int* __restrict__ valid,
                                              const float* __restrict__ W1,
                                              const float* __restrict__ b1,
                                              const float* __restrict__ W2,
                                              const float* __restrict__ b2,
                                              const float* __restrict__ deg,
                                              float* __restrict__ out,
                                              int m) {
    __shared__ __align__(16) float lds1[CH * CH];
    __shared__ __align__(16) float lds2[CH * CH];
    __shared__ float lb1[CH];
    __shared__ float lb2[CH];
    for (int i = threadIdx.x; i < CH * CH; i += 256) {
        // W1/W2 are [j][k] row-major; B[k][n=j] => swap index roles.
        const int d = bfrag_idx(i & 63, i >> 6);
        lds1[d] = W1[i];
        lds2[d] = W2[i];
    }
    if (threadIdx.x < CH) {
        lb1[threadIdx.x] = b1[threadIdx.x];
        lb2[threadIdx.x] = b2[threadIdx.x];
    }
    __syncthreads();

    const int wave = threadIdx.x >> 5;
    const int lane = threadIdx.x & 31;
    const int hi   = lane >> 4;
    const int ln   = lane & 15;

    const int rowBase = (blockIdx.x * 8 + wave) * 16;
    if (rowBase >= m) return;         // wave-uniform exit

    int row = rowBase + ln;
    if (row >= m) row = m - 1;        // clamp (loads only)
    const int src = valid[row];
    const float* arow = X + (size_t)src * CH + 2 * hi;
    __builtin_prefetch(arow, 0, 0);

    const float* b1base = lds1 + (lane << 1);
    const float* b2base = lds2 + (lane << 1);

    v8f acc1[4] = {};
    v8f acc2[4] = {};
#pragma unroll
    for (int s = 0; s < 16; ++s) {
        v2f a;
        a.x = arow[4 * s + 0];
        a.y = arow[4 * s + 1];
#pragma unroll
        for (int t = 0; t < 4; ++t) {
            const int off = (s * 4 + t) << 6;
            const v2f bA = *(const v2f*)(b1base + off);
            acc1[t] = __builtin_amdgcn_wmma_f32_16x16x4_f32(
                false, a, false, bA, (short)0, acc1[t], false, false);
            const v2f bB = *(const v2f*)(b2base + off);
            acc2[t] = __builtin_amdgcn_wmma_f32_16x16x4_f32(
                false, a, false, bB, (short)0, acc2[t], false, false);
        }
    }
    // Fused epilogue in the C/D register layout.
#pragma unroll
    for (int t = 0; t < 4; ++t) {
        const int n = t * 16 + ln;
        const float bb1 = lb1[n];
        const float bb2 = lb2[n];
#pragma unroll
        for (int r = 0; r < 8; ++r) {
            const int orow = rowBase + r + 8 * hi;
            if (orow < m) {
                const float d = deg[orow];
                const size_t oi = (size_t)orow * CH + n;
                out[oi] = d * (acc1[t][r] + bb1) + out[oi] + acc2[t][r] + bb2;
            }
        }
    }
}

// ---------------------------------------------------------------------------
// Host-side launch
// ---------------------------------------------------------------------------
extern "C" void kernel_launch(void* const* d_in, const int* in_sizes, int n_in,
                              void* d_out, int out_size, void* d_ws, size_t ws_size,
                              hipStream_t stream) {
    const float* X      = (const float*)d_in[0];  // [N,64]
    const int*   valid  = (const int*)  d_in[1];  // [M]
    const int*   index  = (const int*)  d_in[2];  // [E]
    const int*   index1 = (const int*)  d_in[3];  // [E]
    const float* ew     = (const float*)d_in[4];  // [E]
    const float* weight = (const float*)d_in[5];  // [64,64]
    const float* W1     = (const float*)d_in[6];  // [64,64]
    const float* b1     = (const float*)d_in[7];  // [64]
    const float* W2     = (const float*)d_in[8];  // [64,64]
    const float* b2     = (const float*)d_in[9];  // [64]

    const int N = in_sizes[0] / CH;
    const int M = in_sizes[1];
    const int E = in_sizes[2];

    float* out  = (float*)d_out;                  // [M,64], used as aggr acc
    // Workspace layout: deg [M] | emb_w [N,64]  (256-byte aligned split)
    char*  ws   = (char*)d_ws;
    float* deg  = (float*)ws;
    size_t degBytes = ((size_t)M * sizeof(float) + 255) & ~(size_t)255;
    float* embw = (float*)(ws + degBytes);

    // 0) zero aggr + deg
    {
        int blocks = (M * CH + 255) / 256;
        kzero<<<blocks, 256, 0, stream>>>(out, deg, M);
    }
    // 1) emb_w = X @ weight
    {
        int blocks = (N + 127) / 128;             // 8 waves x 16 rows / block
        kproj<<<blocks, 256, 0, stream>>>(X, weight, embw, N);
    }
    // 2) edge scatter (aggr + deg)
    {
        long long threads = (long long)E * 16;
        int blocks = (int)((threads + 255) / 256);
        kscatter<<<blocks, 256, 0, stream>>>(index, index1, ew, embw, out, deg, E);
    }
    // 3) fused epilogue GEMMs + combine
    {
        int blocks = (M + 127) / 128;
        kfinal<<<blocks, 256, 0, stream>>>(X, valid, W1, b1, W2, b2, deg, out, M);
    }
}